// DiffMMM_51625506898618
// MI455X (gfx1250) — compile-verified
//
#include <hip/hip_runtime.h>
#include <hip/hip_bf16.h>
#include <math.h>

// ---------------------------------------------------------------------------
// DiffMMM forward for MI455X (gfx1250, wave32).
//  k1: hill transform of media (T x 4)
//  k2: 2-layer MLP + heads via V_WMMA_F32_16X16X4_F32, per-16-segment strip
//  k3: sequential scan; 32 independent single-wave blocks (32 segments each),
//      state in registers, shfl_xor wave reduction, no barriers in hot loop
//  k4: deterministic fixed-order combine of the 32 per-block partial sums
// ---------------------------------------------------------------------------

typedef __attribute__((ext_vector_type(2))) float v2f;
typedef __attribute__((ext_vector_type(8))) float v8f;

#define T_STEPS 8192
#define K_SEG   1024
#define F_DIM   64
#define H_DIM   256

__device__ __forceinline__ float sig_fast(float x) {
    // v_exp_f32 + v_rcp_f32 (hot loop)
    return __builtin_amdgcn_rcpf(1.0f + __expf(-x));
}
__device__ __forceinline__ float sig_acc(float x) {
    return 1.0f / (1.0f + expf(-x));
}
__device__ __forceinline__ float softplus_acc(float x) {
    return fmaxf(x, 0.0f) + log1pf(expf(-fabsf(x)));
}

// ---------------- Kernel 1: hill transform -------------------------------
__global__ void hill_kernel(const float* __restrict__ xm,
                            const float* __restrict__ gamma,
                            const float* __restrict__ kappa,
                            float* __restrict__ media) {
    int id = blockIdx.x * blockDim.x + threadIdx.x;
    if (id >= T_STEPS * 4) return;
    int t = id >> 2, c = id & 3;       // only channels 0..3 are used downstream
    float g = fabsf(gamma[c]);
    float k = fabsf(kappa[c]);
    float x = xm[t * 5 + c];
    float xp = __powf(x + 1e-8f, g);
    float kp = __powf(k, g);
    media[t * 4 + c] = xp / (xp + kp);
}

// ---------------- Kernel 2: MLP + heads (WMMA f32 16x16x4) ---------------
// One block = one 16-segment strip. 256 threads = 8 waves, each wave owns
// two 16x16 output column tiles of the 16x256 activation strip.
#define SA_STR 68    // 64 + 4 pad (avoid LDS bank conflicts on K-striped reads)
#define H_STR  260   // 256 + 4 pad

__global__ __launch_bounds__(256)
void mlp_heads_kernel(const float* __restrict__ S,
                      const float* __restrict__ W1, const float* __restrict__ b1,
                      const float* __restrict__ W2, const float* __restrict__ b2,
                      const float* __restrict__ Wa_ua, const float* __restrict__ ba_ua,
                      const float* __restrict__ Wb_ua, const float* __restrict__ bb_ua,
                      const float* __restrict__ Wl,    const float* __restrict__ bl,
                      const float* __restrict__ Wa_ac, const float* __restrict__ ba_ac,
                      const float* __restrict__ Wb_ac, const float* __restrict__ bb_ac,
                      const float* __restrict__ Wb_acd,const float* __restrict__ bb_acd,
                      float* __restrict__ segP) {
    __shared__ float sA[16 * SA_STR];   // seg_attr strip  (16 x 64)
    __shared__ float h1[16 * H_STR];    // relu(S@W1+b1)   (16 x 256)
    __shared__ float h2[16 * H_STR];    // relu(h1@W2+b2)  (16 x 256)

    const int strip = blockIdx.x;       // 0..63
    const int tid   = threadIdx.x;
    const int lane  = tid & 31;
    const int wave  = tid >> 5;
    const int m     = lane & 15;        // row within half / output column sel
    const int half  = lane >> 4;        // 0: lanes 0-15, 1: lanes 16-31

    // cooperative load of the 16x64 seg_attr strip (row-major (K,F))
    for (int i = tid; i < 16 * F_DIM; i += 256) {
        int r = i >> 6, c = i & 63;
        sA[r * SA_STR + c] = S[(strip * 16 + r) * F_DIM + c];
    }
    __syncthreads();

    // ---- stage 1: h1 = relu(S_strip @ W1 + b1), K-dim = 64 ----
    for (int ct = 0; ct < 2; ++ct) {
        const int col0 = (wave * 2 + ct) * 16;
        const int n    = col0 + m;
        v8f acc = {};
        for (int k0 = 0; k0 < F_DIM; k0 += 4) {
            const int kk = k0 + 2 * half;
            // A 16x4 f32: lanes 0-15 -> K={k0,k0+1}, lanes 16-31 -> K={k0+2,k0+3}
            v2f a = { sA[m * SA_STR + kk], sA[m * SA_STR + kk + 1] };
            // B 4x16 f32 (mirrored layout): v0/v1 = rows kk, kk+1 at column n
            v2f b = { W1[kk * H_DIM + n], W1[(kk + 1) * H_DIM + n] };
            acc = __builtin_amdgcn_wmma_f32_16x16x4_f32(
                false, a, false, b, (short)0, acc, false, false);
        }
        const float bias = b1[n];
        #pragma unroll
        for (int r = 0; r < 8; ++r) {
            const int row = r + 8 * half;     // C/D layout: vgpr r -> M=r / r+8
            float v = acc[r] + bias;
            h1[row * H_STR + n] = v > 0.0f ? v : 0.0f;
        }
    }
    __syncthreads();

    // ---- stage 2: h2 = relu(h1 @ W2 + b2), K-dim = 256 ----
    for (int ct = 0; ct < 2; ++ct) {
        const int col0 = (wave * 2 + ct) * 16;
        const int n    = col0 + m;
        v8f acc = {};
        for (int k0 = 0; k0 < H_DIM; k0 += 4) {
            const int kk = k0 + 2 * half;
            v2f a = { h1[m * H_STR + kk], h1[m * H_STR + kk + 1] };
            v2f b = { W2[kk * H_DIM + n], W2[(kk + 1) * H_DIM + n] };
            acc = __builtin_amdgcn_wmma_f32_16x16x4_f32(
                false, a, false, b, (short)0, acc, false, false);
        }
        const float bias = b2[n];
        #pragma unroll
        for (int r = 0; r < 8; ++r) {
            const int row = r + 8 * half;
            float v = acc[r] + bias;
            h2[row * H_STR + n] = v > 0.0f ? v : 0.0f;
        }
    }
    __syncthreads();

    // ---- stage 3: 9 linear heads per segment row ----
    // segP per k (stride 12): [0]=aUA [1..3]=bUA0..2 [4]=probAU [5]=aAC
    //                         [6]=bACd [7]=bAC0 [8]=bAC1 [9]=sig(aUA) [10]=sig(aAC)
    if (tid < 16 * 9) {
        const int row  = tid / 9;
        const int head = tid % 9;
        const float* w; int stride; float bias;
        switch (head) {
            case 0:  w = Wa_ua;              stride = 1; bias = ba_ua[0];      break;
            case 1: case 2: case 3:
                     w = Wb_ua + (head - 1); stride = 3; bias = bb_ua[head-1]; break;
            case 4:  w = Wl;                 stride = 1; bias = bl[0];         break;
            case 5:  w = Wa_ac;              stride = 1; bias = ba_ac[0];      break;
            case 6:  w = Wb_acd;             stride = 1; bias = bb_acd[0];     break;
            default: w = Wb_ac + (head - 7); stride = 2; bias = bb_ac[head-7]; break;
        }
        float d = bias;
        for (int j = 0; j < H_DIM; ++j)
            d += h2[row * H_STR + j] * w[j * stride];

        float* P = segP + (strip * 16 + row) * 12;
        switch (head) {
            case 0: P[0] = d; P[9]  = sig_acc(d);      break;
            case 1: P[1] = softplus_acc(d);            break;
            case 2: P[2] = softplus_acc(d);            break;
            case 3: P[3] = softplus_acc(d);            break;
            case 4: P[4] = sig_acc(d);                 break;
            case 5: P[5] = d; P[10] = sig_acc(d);      break;
            case 6: P[6] = softplus_acc(d);            break;
            case 7: P[7] = softplus_acc(d);            break;
            case 8: P[8] = softplus_acc(d);            break;
        }
    }
}

// ---------------- Kernel 3: sequential scan ------------------------------
// 32 blocks x 32 lanes; block b owns segments b*32..b*32+31. State (u,a) in
// registers; per step: 6 sigmoids, flow update, wave shfl_xor reduction,
// lane 0 stores the block partial. No LDS/barriers in the loop.
__global__ __launch_bounds__(32)
void scan_kernel(const float* __restrict__ media,
                 const float* __restrict__ segP,
                 float* __restrict__ partial) {
    const int lane = threadIdx.x;
    const int k    = blockIdx.x * 32 + lane;
    const float* P = segP + k * 12;
    const float aUA = P[0], bU0 = P[1], bU1 = P[2], bU2 = P[3];
    const float pAU = P[4], aAC = P[5], bAd = P[6], bA0 = P[7], bA1 = P[8];
    const float base2_ua = 2.0f * P[9];
    const float base2_ac = 2.0f * P[10];

    const float M = 100000.0f / (float)K_SEG;   // 97.65625
    float u = 0.85f * M;
    float a = 0.14f * M;

    const float4* mp = (const float4*)media;
    float4 pm = mp[1];                           // software pipeline: 1 ahead
    for (int t = 1; t < T_STEPS; ++t) {
        const int tn = (t + 1 < T_STEPS) ? (t + 1) : t;
        float4 pm_next = mp[tn];

        const float tv = pm.x, dp = pm.y, gn = pm.z, br = pm.w;
        float pua = sig_fast(aUA + bU0 * tv) + sig_fast(aUA + bU1 * dp)
                  + sig_fast(aUA + bU2 * gn) - base2_ua;
        float pac = sig_fast(aAC + bAd * dp) + sig_fast(aAC + bA0 * gn)
                  + sig_fast(aAC + bA1 * br) - base2_ac;

        float fUA = u * pua;
        float fAU = a * pAU;
        float fAC = a * pac;
        float u2 = u - fUA + fAU; u = u2 > 0.0f ? u2 : 0.0f;
        float a2 = a + fUA - fAU - fAC; a = a2 > 0.0f ? a2 : 0.0f;

        float s = fAC;                            // wave32 reduction
        s += __shfl_xor(s, 1,  32);
        s += __shfl_xor(s, 2,  32);
        s += __shfl_xor(s, 4,  32);
        s += __shfl_xor(s, 8,  32);
        s += __shfl_xor(s, 16, 32);
        if (lane == 0) partial[blockIdx.x * T_STEPS + t] = s;

        pm = pm_next;
    }
}

// ---------------- Kernel 4: deterministic combine ------------------------
__global__ void finalize_kernel(const float* __restrict__ partial,
                                const float* __restrict__ base_conv,
                                float* __restrict__ out) {
    int t = blockIdx.x * blockDim.x + threadIdx.x;
    if (t >= T_STEPS) return;
    float base = base_conv[0];
    if (t == 0) {                                 // c0*K = 0.01*(100000/K)*K = 1000
        out[0] = base + 1000.0f;
        return;
    }
    float s = 0.0f;
    #pragma unroll
    for (int b = 0; b < 32; ++b) s += partial[b * T_STEPS + t];
    out[t] = base + s;
}

// ---------------------------------------------------------------------------
extern "C" void kernel_launch(void* const* d_in, const int* in_sizes, int n_in,
                              void* d_out, int out_size, void* d_ws, size_t ws_size,
                              hipStream_t stream) {
    // setup_inputs() dict order (params dict flattened in insertion order)
    const float* x_media = (const float*)d_in[0];   // (8192,5)
    const float* seg     = (const float*)d_in[1];   // (1024,64)
    const float* gamma   = (const float*)d_in[2];   // (5,)
    const float* kappa   = (const float*)d_in[3];   // (5,)
    const float* W1      = (const float*)d_in[4];   // (64,256)
    const float* b1      = (const float*)d_in[5];   // (256,)
    const float* W2      = (const float*)d_in[6];   // (256,256)
    const float* b2      = (const float*)d_in[7];   // (256,)
    const float* Wa_ua   = (const float*)d_in[8];   // (256,1)
    const float* ba_ua   = (const float*)d_in[9];   // (1,)
    const float* Wb_ua   = (const float*)d_in[10];  // (256,3)
    const float* bb_ua   = (const float*)d_in[11];  // (3,)
    const float* Wl      = (const float*)d_in[12];  // (256,1)
    const float* bl      = (const float*)d_in[13];  // (1,)
    const float* Wa_ac   = (const float*)d_in[14];  // (256,1)
    const float* ba_ac   = (const float*)d_in[15];  // (1,)
    const float* Wb_ac   = (const float*)d_in[16];  // (256,2)
    const float* bb_ac   = (const float*)d_in[17];  // (2,)
    const float* Wb_acd  = (const float*)d_in[18];  // (256,1)
    const float* bb_acd  = (const float*)d_in[19];  // (1,)
    const float* base_c  = (const float*)d_in[20];  // scalar
    float* out = (float*)d_out;

    char* ws = (char*)d_ws;
    float* media   = (float*)ws;                                   // T*4 f32
    float* segP    = (float*)(ws + (size_t)T_STEPS * 4 * 4);       // K*12 f32
    float* partial = (float*)(ws + (size_t)T_STEPS * 4 * 4
                                 + (size_t)K_SEG * 12 * 4);        // 32*T f32

    hipLaunchKernelGGL(hill_kernel, dim3((T_STEPS * 4 + 255) / 256), dim3(256),
                       0, stream, x_media, gamma, kappa, media);

    hipLaunchKernelGGL(mlp_heads_kernel, dim3(K_SEG / 16), dim3(256),
                       0, stream, seg, W1, b1, W2, b2,
                       Wa_ua, ba_ua, Wb_ua, bb_ua, Wl, bl,
                       Wa_ac, ba_ac, Wb_ac, bb_ac, Wb_acd, bb_acd, segP);

    hipLaunchKernelGGL(scan_kernel, dim3(32), dim3(32),
                       0, stream, media, segP, partial);

    hipLaunchKernelGGL(finalize_kernel, dim3((T_STEPS + 255) / 256), dim3(256),
                       0, stream, partial, base_c, out);
}